// GCNEncoder_82781199663549
// MI455X (gfx1250) — compile-verified
//
#include <hip/hip_runtime.h>

typedef __attribute__((ext_vector_type(2))) float v2f;
typedef __attribute__((ext_vector_type(8))) float v8f;

#define LEAKY 0.01f
#define EPS_BN 1e-5f

// ---------------- degree / dinv ----------------
__global__ void k_init_deg(float* __restrict__ deg, int n) {
  int i = blockIdx.x * blockDim.x + threadIdx.x;
  if (i < n) deg[i] = 1.0f;  // self-loop
}

__global__ void k_edge_deg(const int* __restrict__ dst, float* __restrict__ deg, int e) {
  int i = blockIdx.x * blockDim.x + threadIdx.x;
  if (i < e) atomicAdd(&deg[dst[i]], 1.0f);
}

__global__ void k_dinv(float* __restrict__ deg, int n) {
  int i = blockIdx.x * blockDim.x + threadIdx.x;
  if (i < n) deg[i] = rsqrtf(deg[i]);
}

// ---------------- fp32 WMMA GEMM: C[M,NOUT] = A[M,K] @ B[K,NOUT] ----------------
// One wave computes one 16x16 output tile via V_WMMA_F32_16X16X4_F32.
// Per-lane layout (ISA 7.12.2): A: row=lane&15, k = 2*(lane>>4)+{0,1}
//                               B: col=lane&15, k = 2*(lane>>4)+{0,1}
//                               C/D reg r: row = r + 8*(lane>>4), col = lane&15
template<int K, int NOUT>
__global__ void k_gemm_wmma(const float* __restrict__ A, const float* __restrict__ B,
                            float* __restrict__ C, int M) {
  const int lane = threadIdx.x & 31;
  const int wave = threadIdx.x >> 5;
  const int wavesPerBlock = blockDim.x >> 5;
  const int tilesN = NOUT / 16;
  const int tilesM = M / 16;  // M = 50000 is an exact multiple of 16
  int tile = blockIdx.x * wavesPerBlock + wave;
  if (tile >= tilesM * tilesN) return;  // wave-uniform; surviving waves keep EXEC all-1s
  int tm = tile / tilesN;
  int tn = tile % tilesN;

  const int mi = lane & 15;
  const int kh = lane >> 4;  // 0 or 1
  const float* arow = A + (size_t)(tm * 16 + mi) * K;
  const float* bcol = B + tn * 16 + mi;

  v8f acc = {};
#pragma unroll
  for (int k0 = 0; k0 < K; k0 += 4) {
    int kk = k0 + 2 * kh;
    v2f a, b;
    a.x = arow[kk];
    a.y = arow[kk + 1];
    b.x = bcol[(size_t)kk * NOUT];
    b.y = bcol[(size_t)(kk + 1) * NOUT];
    acc = __builtin_amdgcn_wmma_f32_16x16x4_f32(false, a, false, b,
                                                (short)0, acc, false, false);
  }

  float* crow = C + (size_t)(tm * 16 + 8 * kh) * NOUT + tn * 16 + mi;
#pragma unroll
  for (int r = 0; r < 8; ++r)
    crow[(size_t)r * NOUT] = acc[r];
}

// ---------------- agg init: bias + self-loop term (no atomics) ----------------
template<int D>
__global__ void k_init_agg(const float* __restrict__ h, const float* __restrict__ dinv,
                           const float* __restrict__ bias, float* __restrict__ agg, int n) {
  size_t i = (size_t)blockIdx.x * blockDim.x + threadIdx.x;
  if (i >= (size_t)n * D) return;
  int row = (int)(i / D);
  int c = (int)(i % D);
  float w = dinv[row];
  agg[i] = bias[c] + h[i] * (w * w);
}

// ---------------- edge scatter: one wave per edge, lanes split channels ----------------
template<int D>
__global__ void k_scatter(const int* __restrict__ src, const int* __restrict__ dst,
                          const float* __restrict__ dinv, const float* __restrict__ h,
                          float* __restrict__ agg, int e) {
  int w = (int)(((size_t)blockIdx.x * blockDim.x + threadIdx.x) >> 5);
  int lane = threadIdx.x & 31;
  if (w >= e) return;
  int s = src[w], d = dst[w];
  float nw = dinv[s] * dinv[d];
  constexpr int VEC = D / 32;  // 4 for D=128, 2 for D=64
  const float* hp = h + (size_t)s * D + lane * VEC;
  float* ap = agg + (size_t)d * D + lane * VEC;
  if constexpr (VEC == 4) {
    float4 v = *(const float4*)hp;
    atomicAdd(ap + 0, v.x * nw);
    atomicAdd(ap + 1, v.y * nw);
    atomicAdd(ap + 2, v.z * nw);
    atomicAdd(ap + 3, v.w * nw);
  } else {
    float2 v = *(const float2*)hp;
    atomicAdd(ap + 0, v.x * nw);
    atomicAdd(ap + 1, v.y * nw);
  }
}

// ---------------- BatchNorm ----------------
__global__ void k_zero(float* __restrict__ p, int n) {
  int i = blockIdx.x * blockDim.x + threadIdx.x;
  if (i < n) p[i] = 0.0f;
}

// stats[0..D) = sum, stats[D..2D) = sumsq
template<int D>
__global__ void k_bn_stats(const float* __restrict__ x, float* __restrict__ stats, int n) {
  const int rowsPerBlk = blockDim.x / D;
  int c = threadIdx.x % D;
  int r = blockIdx.x * rowsPerBlk + threadIdx.x / D;
  int stride = gridDim.x * rowsPerBlk;
  float s = 0.f, ss = 0.f;
  for (; r < n; r += stride) {
    float v = x[(size_t)r * D + c];
    s += v;
    ss += v * v;
  }
  atomicAdd(&stats[c], s);
  atomicAdd(&stats[D + c], ss);
}

// stats[2D..3D) = scale, stats[3D..4D) = shift
template<int D>
__global__ void k_bn_finalize(float* __restrict__ stats, const float* __restrict__ gamma,
                              const float* __restrict__ beta, int n) {
  int c = threadIdx.x;
  if (c < D) {
    float invn = 1.0f / (float)n;
    float mean = stats[c] * invn;
    float var = stats[D + c] * invn - mean * mean;  // biased variance
    float sc = gamma[c] * rsqrtf(var + EPS_BN);
    stats[2 * D + c] = sc;
    stats[3 * D + c] = beta[c] - mean * sc;
  }
}

template<int D>
__global__ void k_bn_apply(const float* __restrict__ x, const float* __restrict__ stats,
                           float* __restrict__ y, int n) {
  size_t i = (size_t)blockIdx.x * blockDim.x + threadIdx.x;
  if (i >= (size_t)n * D) return;
  int c = (int)(i % D);
  float v = x[i] * stats[2 * D + c] + stats[3 * D + c];
  y[i] = v > 0.0f ? v : LEAKY * v;
}

// ---------------- launch ----------------
extern "C" void kernel_launch(void* const* d_in, const int* in_sizes, int n_in,
                              void* d_out, int out_size, void* d_ws, size_t ws_size,
                              hipStream_t stream) {
  const float* x   = (const float*)d_in[0];
  const int*   ei  = (const int*)d_in[1];
  const float* W1  = (const float*)d_in[2];
  const float* b1  = (const float*)d_in[3];
  const float* g1  = (const float*)d_in[4];
  const float* be1 = (const float*)d_in[5];
  const float* W2  = (const float*)d_in[6];
  const float* b2  = (const float*)d_in[7];
  const float* g2  = (const float*)d_in[8];
  const float* be2 = (const float*)d_in[9];
  const float* W3  = (const float*)d_in[10];
  const float* b3  = (const float*)d_in[11];
  float* out = (float*)d_out;

  const int N = in_sizes[0] / 64;  // 50000
  const int E = in_sizes[1] / 2;   // 800000
  const int* src = ei;
  const int* dst = ei + E;

  // workspace layout (floats), 256B-aligned regions
  float* dinv  = (float*)d_ws;                       // N
  size_t o = ((size_t)N * 4 + 255) & ~(size_t)255;
  float* bufH  = (float*)((char*)d_ws + o);          // N*128
  float* bufA  = bufH + (size_t)N * 128;             // N*128
  float* stats = bufA + (size_t)N * 128;             // 4*128

  const int TB = 256;
  const int nBlkN   = (N + TB - 1) / TB;
  const int nBlkE   = (E + TB - 1) / TB;
  const int nBlkEdgeWave = (E * 32 + TB - 1) / TB;   // one wave per edge
  const int nBlk128 = ((size_t)N * 128 + TB - 1) / TB;
  const int nBlk64  = ((size_t)N * 64 + TB - 1) / TB;

  // degrees -> dinv
  k_init_deg<<<nBlkN, TB, 0, stream>>>(dinv, N);
  k_edge_deg<<<nBlkE, TB, 0, stream>>>(dst, dinv, E);
  k_dinv<<<nBlkN, TB, 0, stream>>>(dinv, N);

  // ---- layer 1: x @ W1 -> scatter -> BN -> leaky ----
  {
    int tiles = (N / 16) * (128 / 16);
    k_gemm_wmma<64, 128><<<(tiles + 3) / 4, 128, 0, stream>>>(x, W1, bufH, N);
  }
  k_init_agg<128><<<nBlk128, TB, 0, stream>>>(bufH, dinv, b1, bufA, N);
  k_scatter<128><<<nBlkEdgeWave, TB, 0, stream>>>(src, dst, dinv, bufH, bufA, E);
  k_zero<<<1, 256, 0, stream>>>(stats, 256);
  k_bn_stats<128><<<256, TB, 0, stream>>>(bufA, stats, N);
  k_bn_finalize<128><<<1, 128, 0, stream>>>(stats, g1, be1, N);
  k_bn_apply<128><<<nBlk128, TB, 0, stream>>>(bufA, stats, bufH, N);  // z1 -> bufH

  // ---- layer 2: z1 @ W2 -> scatter -> BN -> leaky ----
  {
    int tiles = (N / 16) * (64 / 16);
    k_gemm_wmma<128, 64><<<(tiles + 3) / 4, 128, 0, stream>>>(bufH, W2, bufA, N);  // h2 -> bufA
  }
  k_init_agg<64><<<nBlk64, TB, 0, stream>>>(bufA, dinv, b2, bufH, N);
  k_scatter<64><<<nBlkEdgeWave, TB, 0, stream>>>(src, dst, dinv, bufA, bufH, E);
  k_zero<<<1, 256, 0, stream>>>(stats, 256);
  k_bn_stats<64><<<256, TB, 0, stream>>>(bufH, stats, N);
  k_bn_finalize<64><<<1, 64, 0, stream>>>(stats, g2, be2, N);
  k_bn_apply<64><<<nBlk64, TB, 0, stream>>>(bufH, stats, bufA, N);  // z2 -> bufA

  // ---- layer 3: z2 @ W3 -> scatter (+bias) -> d_out ----
  {
    int tiles = (N / 16) * (64 / 16);
    k_gemm_wmma<64, 64><<<(tiles + 3) / 4, 128, 0, stream>>>(bufA, W3, bufH, N);  // h3 -> bufH
  }
  k_init_agg<64><<<nBlk64, TB, 0, stream>>>(bufH, dinv, b3, out, N);
  k_scatter<64><<<nBlkEdgeWave, TB, 0, stream>>>(src, dst, dinv, bufH, out, E);
}